// GAT_70909910057105
// MI455X (gfx1250) — compile-verified
//
#include <hip/hip_runtime.h>
#include <hip/hip_bf16.h>

// ---------------------------------------------------------------------------
// 2-layer GAT, MI455X (gfx1250, wave32).
// N=1024, B=4, C=256, H=4, D=64, E=32768.
//
// Roofline: ~1.5 GFLOP vs ~40 MB of HBM traffic -> memory bound. FP32 kept
// end-to-end; node projections use V_WMMA_F32_16X16X4_F32. Attention is
// target-centric over a deterministic CSR (built once, no atomics anywhere),
// fusing score/exp/denominator/aggregate into one register-resident pass:
//   out = (sum_e exp_e * h_src) / (sum_e exp_e + 1e-16)
// ---------------------------------------------------------------------------

#define GAT_N 1024
#define GAT_B 4
#define GAT_C 256
#define GAT_H 4
#define GAT_D 64
#define GAT_E 32768
#define GAT_CHUNKS (GAT_E / 32)   // 1024 wave-sized chunks for CSR build

typedef float v2f __attribute__((ext_vector_type(2)));
typedef float v4f __attribute__((ext_vector_type(4)));
typedef float v8f __attribute__((ext_vector_type(8)));

__device__ __forceinline__ float dot4(v4f a, v4f b) {
  return a.x * b.x + a.y * b.y + a.z * b.z + a.w * b.w;
}

// ---------------------------------------------------------------------------
// wsum[l*4+h][c] = sum_d We_l[h*64+d, c]   (8 x 256 floats)
// Exact reassociation of the reference's (e_feats @ We.T).sum(-1).
// ---------------------------------------------------------------------------
__global__ __launch_bounds__(256) void gat_wsum_kernel(
    const float* __restrict__ We1, const float* __restrict__ We2,
    float* __restrict__ wsum) {
  int t = blockIdx.x * 256 + threadIdx.x;       // 0..2047
  int l = t >> 10;
  int h = (t >> 8) & 3;
  int c = t & 255;
  const float* W = l ? We2 : We1;
  float s = 0.f;
  #pragma unroll 8
  for (int d = 0; d < GAT_D; ++d) s += W[(h * GAT_D + d) * GAT_C + c];
  wsum[t] = s;                                   // t == (l*4+h)*256 + c
}

// ---------------------------------------------------------------------------
// pe_l[e,h] = a_e_l[h] * dot(edge_features[src[e],trg[e],:], wsum_l[h,:])
// One wave per edge; streams the 32MB gather from the 1GiB tensor exactly
// once (b128 loads), producing pe for BOTH layers.
// ---------------------------------------------------------------------------
__global__ __launch_bounds__(256) void gat_pe_kernel(
    const float* __restrict__ EF,                // [N,N,C]
    const int* __restrict__ src, const int* __restrict__ trg,
    const float* __restrict__ wsum,              // [8,256]
    const float* __restrict__ ae1, const float* __restrict__ ae2,
    float* __restrict__ pe1, float* __restrict__ pe2) {
  __shared__ float ws[8 * GAT_C];
  for (int i = threadIdx.x; i < 8 * GAT_C; i += 256) ws[i] = wsum[i];
  __syncthreads();

  int wave = threadIdx.x >> 5;
  int lane = threadIdx.x & 31;
  int e = blockIdx.x * 8 + wave;

  size_t base = ((size_t)src[e] * GAT_N + (size_t)trg[e]) * GAT_C;
  const v4f* ef4 = (const v4f*)(EF + base) + lane * 2;   // c = 8*lane + j
  v4f v0 = ef4[0], v1 = ef4[1];
  int c0 = lane * 8;

  float p[8];
  #pragma unroll
  for (int j = 0; j < 8; ++j) {
    const float* w = ws + j * GAT_C + c0;
    p[j] = v0.x * w[0] + v0.y * w[1] + v0.z * w[2] + v0.w * w[3]
         + v1.x * w[4] + v1.y * w[5] + v1.z * w[6] + v1.w * w[7];
  }
  #pragma unroll
  for (int j = 0; j < 8; ++j) {
    #pragma unroll
    for (int off = 16; off; off >>= 1) p[j] += __shfl_xor(p[j], off);
  }
  if (lane == 0) {
    pe1[e * 4 + 0] = ae1[0] * p[0];
    pe1[e * 4 + 1] = ae1[1] * p[1];
    pe1[e * 4 + 2] = ae1[2] * p[2];
    pe1[e * 4 + 3] = ae1[3] * p[3];
    pe2[e * 4 + 0] = ae2[0] * p[4];
    pe2[e * 4 + 1] = ae2[1] * p[5];
    pe2[e * 4 + 2] = ae2[2] * p[6];
    pe2[e * 4 + 3] = ae2[3] * p[7];
  }
}

// ---------------------------------------------------------------------------
// Deterministic CSR build over trg (no atomics).
// Step 1: per wave-chunk of 32 edges, compute each edge's rank within its
//         target bucket inside the chunk, and the per-chunk bucket counts.
// ---------------------------------------------------------------------------
__global__ __launch_bounds__(256) void csr_count_kernel(
    const int* __restrict__ trg,
    int* __restrict__ cnt,        // [CHUNKS, N] (pre-zeroed)
    int* __restrict__ rank) {     // [E]
  int wave = threadIdx.x >> 5;
  int lane = threadIdx.x & 31;
  int chunk = blockIdx.x * 8 + wave;            // 1024 chunks
  int e = chunk * 32 + lane;
  int t = trg[e];
  int total = 0, r = 0;
  #pragma unroll
  for (int j = 0; j < 32; ++j) {
    int tj = __shfl(t, j);
    int same = (tj == t) ? 1 : 0;
    total += same;
    if (j < lane) r += same;
  }
  rank[e] = r;
  if (r == 0) cnt[chunk * GAT_N + t] = total;   // bucket leader writes count
}

// ---------------------------------------------------------------------------
// Step 2: single block of 1024 threads (thread == target node):
//   total[n] = sum_ch cnt[ch][n];  rowstart = exclusive scan(total);
//   cnt[ch][n] := rowstart[n] + prefix of earlier chunks (global offsets).
// ---------------------------------------------------------------------------
__global__ __launch_bounds__(1024) void csr_scan_kernel(
    int* __restrict__ cnt,        // [CHUNKS, N] -> rewritten to offsets
    int* __restrict__ rowstart) { // [N+1]
  __shared__ int sdata[GAT_N];
  int n = threadIdx.x;
  int total = 0;
  for (int ch = 0; ch < GAT_CHUNKS; ++ch) total += cnt[ch * GAT_N + n];
  sdata[n] = total;
  __syncthreads();
  #pragma unroll
  for (int off = 1; off < GAT_N; off <<= 1) {   // Hillis-Steele inclusive
    int v = (n >= off) ? sdata[n - off] : 0;
    __syncthreads();
    sdata[n] += v;
    __syncthreads();
  }
  int start = sdata[n] - total;                 // exclusive
  rowstart[n] = start;
  if (n == GAT_N - 1) rowstart[GAT_N] = sdata[n];
  int run = start;
  for (int ch = 0; ch < GAT_CHUNKS; ++ch) {
    int c = cnt[ch * GAT_N + n];
    cnt[ch * GAT_N + n] = run;
    run += c;
  }
}

// Step 3: place each edge at its deterministic CSR position.
__global__ __launch_bounds__(256) void csr_fill_kernel(
    const int* __restrict__ trg, const int* __restrict__ cnt,
    const int* __restrict__ rank, int* __restrict__ eid) {
  int e = blockIdx.x * 256 + threadIdx.x;
  eid[cnt[(e >> 5) * GAT_N + trg[e]] + rank[e]] = e;
}

// ---------------------------------------------------------------------------
// Node projection:  Hout[r, j] = sum_c A[r, c] * W[j, c]     (h = x @ W^T)
// One wave per 16x16 output tile using V_WMMA_F32_16X16X4_F32.
// Fragment layouts per ISA 7.12.2 (32-bit, wave32); EXEC is all-ones.
// ---------------------------------------------------------------------------
__global__ __launch_bounds__(256) void gat_gemm_wmma_kernel(
    const float* __restrict__ A,                 // [4096, 256]
    const float* __restrict__ W,                 // [256, 256]
    float* __restrict__ Hout) {                  // [4096, 256]
  int wave = threadIdx.x >> 5;
  int lane = threadIdx.x & 31;
  int tile = blockIdx.x * 8 + wave;              // 4096 tiles
  int m0 = (tile >> 4) * 16;
  int n0 = (tile & 15) * 16;
  int lm = lane & 15;
  int kh = lane >> 4;                            // 0 or 1

  const float* arow = A + (size_t)(m0 + lm) * GAT_C + 2 * kh;
  const float* brow = W + (size_t)(n0 + lm) * GAT_C + 2 * kh;

  v8f acc = {};
  #pragma unroll 8
  for (int k = 0; k < GAT_C; k += 4) {
    v2f a = *(const v2f*)(arow + k);
    v2f b = *(const v2f*)(brow + k);
    acc = __builtin_amdgcn_wmma_f32_16x16x4_f32(
        false, a, false, b, (short)0, acc, false, false);
  }

  float* o = Hout + (size_t)(m0 + kh * 8) * GAT_C + (n0 + lm);
  #pragma unroll
  for (int v = 0; v < 8; ++v) o[(size_t)v * GAT_C] = acc[v];
}

// ---------------------------------------------------------------------------
// Fused attention: one wave per (target node n, batch b). Lane l owns
// channels c = 8l..8l+7, so head = l>>3 is uniform per lane; the per-edge
// head dot reduces with 3 shfl_xor inside the 8-lane group that exactly
// spans one head. Accumulate unnormalized sums in 8 VGPRs, divide once.
// Zero atomics; every output row written exactly once (deg==0 -> zeros).
// ---------------------------------------------------------------------------
__global__ __launch_bounds__(256) void gat_attn_kernel(
    const float* __restrict__ hbuf,              // [N*B, C]
    const float* __restrict__ pe,                // [E, H]
    const float* __restrict__ a_s, const float* __restrict__ a_t,  // [H*D]
    const int* __restrict__ src,
    const int* __restrict__ eid, const int* __restrict__ rowstart,
    float* __restrict__ out) {                   // [N*B, C]
  int wave = threadIdx.x >> 5;
  int lane = threadIdx.x & 31;
  int idx = blockIdx.x * 8 + wave;               // n*B + b
  int n = idx >> 2, b = idx & 3;
  int h = lane >> 3;                             // this lane's head

  const v4f* as4 = (const v4f*)a_s + lane * 2;
  v4f A0 = as4[0], A1 = as4[1];
  const v4f* at4 = (const v4f*)a_t + lane * 2;
  v4f T0 = at4[0], T1 = at4[1];

  // dot(h_tgt, a_t) for this wave's fixed target row, per head.
  const v4f* ht4 = (const v4f*)(hbuf + (size_t)idx * GAT_C) + lane * 2;
  float td = dot4(ht4[0], T0) + dot4(ht4[1], T1);
  td += __shfl_xor(td, 1); td += __shfl_xor(td, 2); td += __shfl_xor(td, 4);

  int beg = rowstart[n], end = rowstart[n + 1];
  float denom = 0.f;
  v4f acc0 = {0.f, 0.f, 0.f, 0.f}, acc1 = {0.f, 0.f, 0.f, 0.f};

  for (int i = beg; i < end; ++i) {
    int e = eid[i];
    int s = src[e];
    const v4f* hs4 = (const v4f*)(hbuf + ((size_t)s * GAT_B + b) * GAT_C) + lane * 2;
    v4f u0 = hs4[0], u1 = hs4[1];
    float sp = dot4(u0, A0) + dot4(u1, A1);
    sp += __shfl_xor(sp, 1); sp += __shfl_xor(sp, 2); sp += __shfl_xor(sp, 4);
    float sv = sp + td + pe[e * 4 + h];
    sv = sv > 0.f ? sv : 0.2f * sv;              // leaky_relu(0.2)
    float ex = __expf(sv);
    denom += ex;
    acc0 += ex * u0;
    acc1 += ex * u1;
  }

  float inv = 1.f / (denom + 1e-16f);
  v4f* o = (v4f*)(out + (size_t)idx * GAT_C) + lane * 2;
  o[0] = acc0 * inv;
  o[1] = acc1 * inv;
}

// ---------------------------------------------------------------------------
// Host-side launch. Inputs (setup_inputs order):
//  0 x 1 edge_features 2 src 3 trg 4 Wn1 5 We1 6 a_src1 7 a_tgt1 8 a_edge1
//  9 Wn2 10 We2 11 a_src2 12 a_tgt2 13 a_edge2.  Output: x2 [N,B,C] f32.
// ---------------------------------------------------------------------------
extern "C" void kernel_launch(void* const* d_in, const int* in_sizes, int n_in,
                              void* d_out, int out_size, void* d_ws, size_t ws_size,
                              hipStream_t stream) {
  (void)in_sizes; (void)n_in; (void)out_size; (void)ws_size;

  const float* x   = (const float*)d_in[0];
  const float* EF  = (const float*)d_in[1];
  const int*   src = (const int*)d_in[2];
  const int*   trg = (const int*)d_in[3];
  const float* Wn1 = (const float*)d_in[4];
  const float* We1 = (const float*)d_in[5];
  const float* as1 = (const float*)d_in[6];
  const float* at1 = (const float*)d_in[7];
  const float* ae1 = (const float*)d_in[8];
  const float* Wn2 = (const float*)d_in[9];
  const float* We2 = (const float*)d_in[10];
  const float* as2 = (const float*)d_in[11];
  const float* at2 = (const float*)d_in[12];
  const float* ae2 = (const float*)d_in[13];
  float* out = (float*)d_out;

  // Workspace layout (4-byte units)
  float* wsf    = (float*)d_ws;
  float* wsum   = wsf;                                        //  2048
  float* pe1    = wsum + 2048;                                //  E*H
  float* pe2    = pe1  + (size_t)GAT_E * GAT_H;               //  E*H
  float* hbuf   = pe2  + (size_t)GAT_E * GAT_H;               //  N*B*C
  float* x1     = hbuf + (size_t)GAT_N * GAT_B * GAT_C;       //  N*B*C
  int*   cnt    = (int*)(x1 + (size_t)GAT_N * GAT_B * GAT_C); //  CHUNKS*N
  int*   rank   = cnt  + (size_t)GAT_CHUNKS * GAT_N;          //  E
  int*   eid    = rank + GAT_E;                               //  E
  int*   rowst  = eid  + GAT_E;                               //  N+1

  // --- Deterministic CSR over trg (shared by both layers) ---
  hipMemsetAsync(cnt, 0, sizeof(int) * (size_t)GAT_CHUNKS * GAT_N, stream);
  csr_count_kernel<<<GAT_CHUNKS / 8, 256, 0, stream>>>(trg, cnt, rank);
  csr_scan_kernel<<<1, 1024, 0, stream>>>(cnt, rowst);
  csr_fill_kernel<<<GAT_E / 256, 256, 0, stream>>>(trg, cnt, rank, eid);

  // --- Shared edge-feature precompute (streams 32MB once, both layers) ---
  gat_wsum_kernel<<<8, 256, 0, stream>>>(We1, We2, wsum);
  gat_pe_kernel<<<GAT_E / 8, 256, 0, stream>>>(EF, src, trg, wsum, ae1, ae2, pe1, pe2);

  const int gemm_blocks = (GAT_N * GAT_B / 16) * (GAT_C / 16) / 8;  // 512
  const int attn_blocks = GAT_N * GAT_B / 8;                        // 512

  // ---- Layer 1 ----
  gat_gemm_wmma_kernel<<<gemm_blocks, 256, 0, stream>>>(x, Wn1, hbuf);
  gat_attn_kernel<<<attn_blocks, 256, 0, stream>>>(hbuf, pe1, as1, at1, src, eid, rowst, x1);

  // ---- Layer 2 ----
  gat_gemm_wmma_kernel<<<gemm_blocks, 256, 0, stream>>>(x1, Wn2, hbuf);
  gat_attn_kernel<<<attn_blocks, 256, 0, stream>>>(hbuf, pe2, as2, at2, src, eid, rowst, out);
}